// GridPooling_65309272703442
// MI455X (gfx1250) — compile-verified
//
#include <hip/hip_runtime.h>

// GridPooling (unsorted segment max): N=4M points, D=64 feats, S=100K segments.
// Memory-bound scatter-max. HBM floor ~45us (1.06 GB streamed @ 23.3 TB/s).
// Strategy:
//   * 16 lanes/point, float4 (B128) per lane -> wave32 = 2 contiguous rows.
//   * NT temporal hint on the 1 GB signal stream (one-pass, don't thrash L2;
//     keep the 25.6 MB output resident in the 192 MB L2 instead).
//   * Monotone read-filter before each atomic (skips ~90% of RMWs).
//   * Float max via native int atomics: val>=0 -> atomicMax(i32),
//     val<0 -> atomicMin(u32). Single native gfx1250 atomics, no CAS loop,
//     order-independent (bit-deterministic) result.

typedef __attribute__((ext_vector_type(4))) float v4f;

__global__ void gridpool_init_neginf(float* __restrict__ out, int n4) {
  const float ninf = -__builtin_inff();
  v4f v = {ninf, ninf, ninf, ninf};
  v4f* o4 = (v4f*)out;
  int i = blockIdx.x * blockDim.x + threadIdx.x;
  const int stride = gridDim.x * blockDim.x;
  for (; i < n4; i += stride) {
    o4[i] = v;
  }
}

__device__ __forceinline__ void atomic_max_f32_bits(float* addr, float val) {
  // IEEE-754 ordering trick: for val >= 0 the int bit pattern is monotone
  // increasing with the float; for val < 0 the uint pattern is monotone
  // decreasing. Both lower to single non-returning native atomics on gfx1250
  // (global_atomic_max_i32 / global_atomic_min_u32). Memory value is monotone
  // non-decreasing in float order under any interleaving.
  const int vi = __float_as_int(val);
  if (vi >= 0) {
    atomicMax((int*)addr, vi);
  } else {
    atomicMin((unsigned int*)addr, (unsigned int)vi);
  }
}

__global__ void gridpool_segmax(const v4f* __restrict__ signal4,
                                const int* __restrict__ cell_idx,
                                float* __restrict__ out,
                                int n_points) {
  const int tid = blockIdx.x * blockDim.x + threadIdx.x;
  const int chunk = tid & 15;                       // which float4 of a 64-f32 row
  const int pstride = (gridDim.x * blockDim.x) >> 4;
  for (int p = tid >> 4; p < n_points; p += pstride) {
    // All 16 lanes of this point read the same idx (coalesces to one request).
    const int seg = __builtin_nontemporal_load(&cell_idx[p]);
    // Streamed once: non-temporal so the 1 GB signal doesn't evict the
    // L2-resident output array.
    const v4f v = __builtin_nontemporal_load(&signal4[(size_t)p * 16 + chunk]);

    float* o = out + (size_t)seg * 64 + (size_t)(chunk * 4);

    // Monotone filter: output only ever increases, so any observed value is a
    // lower bound on the true current value -> skipping when val <= cur is
    // always safe; staleness only costs a redundant atomic.
    const v4f cur = *(const v4f*)o;
    if (v.x > cur.x) atomic_max_f32_bits(o + 0, v.x);
    if (v.y > cur.y) atomic_max_f32_bits(o + 1, v.y);
    if (v.z > cur.z) atomic_max_f32_bits(o + 2, v.z);
    if (v.w > cur.w) atomic_max_f32_bits(o + 3, v.w);
  }
}

extern "C" void kernel_launch(void* const* d_in, const int* in_sizes, int n_in,
                              void* d_out, int out_size, void* d_ws, size_t ws_size,
                              hipStream_t stream) {
  const float* signal   = (const float*)d_in[0];   // [N, 64] f32
  const int*   cell_idx = (const int*)d_in[1];     // [N] i32
  float*       out      = (float*)d_out;           // [S, 64] f32

  const int n_points = in_sizes[1];                // N
  const int n4       = out_size / 4;               // S*64/4 float4s

  // 1) Re-initialize output to -inf every call (stream-ordered before scatter).
  gridpool_init_neginf<<<4096, 256, 0, stream>>>(out, n4);

  // 2) Scatter-max. 256 threads = 8 wave32s; 16 lanes/point; grid-stride.
  gridpool_segmax<<<16384, 256, 0, stream>>>((const v4f*)signal, cell_idx, out,
                                             n_points);
}